// GraphSAGE_49804440764417
// MI455X (gfx1250) — compile-verified
//
#include <hip/hip_runtime.h>
#include <hip/hip_bf16.h>

// GraphSAGE (2x SAGEConv, mean aggregation) for MI455X / gfx1250.
//
// Cost model: GEMMs are ~5 GFLOP (negligible). Edge aggregation dominates:
// 2 * E * 64 gathered floats + 2 * E * 64 f32 atomic RMWs, all against a
// working set (~110 MB) that fits in the 192 MB L2 -> L2-resident atomic
// throughput problem. So: native global_atomic_add_f32 at device scope,
// coalesced float4 gathers, and f32-precision WMMA (v_wmma_f32_16x16x4_f32)
// for the dense parts so reference semantics are preserved at zero cost.

typedef __attribute__((ext_vector_type(2))) float v2f;
typedef __attribute__((ext_vector_type(8))) float v8f;

#define D_IN 64  // both layers have K = 64

__device__ __forceinline__ void atomic_add_f32_dev(float* p, float v) {
  // Native L2 RMW, no return (STOREcnt), device scope so all WGPs agree.
  asm volatile("global_atomic_add_f32 %0, %1, off scope:SCOPE_DEV"
               :: "v"(p), "v"(v) : "memory");
}

__global__ void fill_zero_kernel(float* __restrict__ p, long n) {
  long i = (long)blockIdx.x * blockDim.x + threadIdx.x;
  long stride = (long)gridDim.x * blockDim.x;
  for (; i < n; i += stride) p[i] = 0.0f;
}

__global__ void degree_kernel(const long long* __restrict__ dst,
                              float* __restrict__ deg, int E) {
  int t = blockIdx.x * blockDim.x + threadIdx.x;
  if (t >= E) return;
  atomic_add_f32_dev(deg + dst[t], 1.0f);
}

// One edge per 16 lanes; each lane moves 4 consecutive features.
// A wave covers 2 edges; the 16 lanes of one edge read one full 64-float
// source row (256B, perfectly coalesced) and issue 4 f32 atomics each.
__global__ void scatter_kernel(const float* __restrict__ feat,
                               const long long* __restrict__ src,
                               const long long* __restrict__ dst,
                               float* __restrict__ agg, int E) {
  long t = (long)blockIdx.x * blockDim.x + threadIdx.x;
  int e = (int)(t >> 4);
  if (e >= E) return;
  int f = ((int)t & 15) << 2;
  long s = (long)src[e];
  long d = (long)dst[e];
  float4 v = *(const float4*)(feat + s * D_IN + f);
  float* p = agg + d * D_IN + f;
  atomic_add_f32_dev(p + 0, v.x);
  atomic_add_f32_dev(p + 1, v.y);
  atomic_add_f32_dev(p + 2, v.z);
  atomic_add_f32_dev(p + 3, v.w);
}

// Fused SAGEConv: out[16-node tile] = (agg/deg) @ W_l + xself @ W_r + bias,
// optional ReLU. One wave32 per 16 nodes; K = 64 (16 chunks of 4) in f32 WMMA.
//
// V_WMMA_F32_16X16X4_F32 VGPR layouts (ISA 7.12.2):
//   A (16x4):  lane l -> row l&15, v0 = K[2*(l>>4)], v1 = K[2*(l>>4)+1]
//   B (4x16):  lane l -> col l&15, v0 = K-row 2*(l>>4), v1 = K-row 2*(l>>4)+1
//   C/D:       VGPR j -> row j + 8*(l>>4), col l&15
__global__ __launch_bounds__(256)
void sage_wmma_kernel(const float* __restrict__ xself,
                      const float* __restrict__ agg,
                      const float* __restrict__ deg,
                      const float* __restrict__ Wl,
                      const float* __restrict__ Wr,
                      const float* __restrict__ bias,
                      float* __restrict__ out,
                      int N, int Dout, int do_relu) {
  const int wave = threadIdx.x >> 5;
  const int lane = threadIdx.x & 31;
  const int node_base = ((blockIdx.x << 3) + wave) << 4;
  if (node_base >= N) return;        // wave-uniform: EXEC stays all-ones (N%16==0)

  const int row = lane & 15;
  const int hi  = lane >> 4;
  const int node = node_base + row;

  // Fold the mean (divide-by-degree) into a row scaling of the A matrix.
  const float inv = 1.0f / fmaxf(deg[node], 1.0f);

  // Hoist all A fragments (both operands, all 16 K-chunks) into registers:
  // 64 VGPRs; reused across every output n-tile.
  v2f a_m[16], a_s[16];
  const float* pm = agg   + (long)node * D_IN + 2 * hi;
  const float* ps = xself + (long)node * D_IN + 2 * hi;
#pragma unroll
  for (int kk = 0; kk < 16; ++kk) {
    v2f am = *(const v2f*)(pm + (kk << 2));
    am.x *= inv; am.y *= inv;
    a_m[kk] = am;
    a_s[kk] = *(const v2f*)(ps + (kk << 2));
  }

  const int ntiles = Dout >> 4;
  for (int nt = 0; nt < ntiles; ++nt) {
    const int ncol = (nt << 4) + row;
    const float bv = bias[ncol];
    v8f acc = { bv, bv, bv, bv, bv, bv, bv, bv };
#pragma unroll
    for (int kk = 0; kk < 16; ++kk) {
      const int k0 = (kk << 2) + 2 * hi;
      v2f bl, br;
      bl.x = Wl[k0 * Dout + ncol];
      bl.y = Wl[(k0 + 1) * Dout + ncol];
      br.x = Wr[k0 * Dout + ncol];
      br.y = Wr[(k0 + 1) * Dout + ncol];
      acc = __builtin_amdgcn_wmma_f32_16x16x4_f32(
          false, a_m[kk], false, bl, (short)0, acc, false, false);
      acc = __builtin_amdgcn_wmma_f32_16x16x4_f32(
          false, a_s[kk], false, br, (short)0, acc, false, false);
    }
#pragma unroll
    for (int j = 0; j < 8; ++j) {
      float v = acc[j];
      if (do_relu) v = fmaxf(v, 0.0f);
      out[(long)(node_base + j + 8 * hi) * Dout + (nt << 4) + row] = v;
    }
  }
}

extern "C" void kernel_launch(void* const* d_in, const int* in_sizes, int n_in,
                              void* d_out, int out_size, void* d_ws, size_t ws_size,
                              hipStream_t stream) {
  const float*     x    = (const float*)d_in[0];
  const long long* edge = (const long long*)d_in[1];   // int64 [2, E]
  const float*     Wl1  = (const float*)d_in[2];
  const float*     Wr1  = (const float*)d_in[3];
  const float*     b1   = (const float*)d_in[4];
  const float*     Wl2  = (const float*)d_in[5];
  const float*     Wr2  = (const float*)d_in[6];
  const float*     b2   = (const float*)d_in[7];
  float*           out  = (float*)d_out;

  const int N = in_sizes[0] / D_IN;     // 100000
  const int E = in_sizes[1] / 2;        // 1200000
  const long long* src = edge;
  const long long* dst = edge + E;

  // Workspace: agg[N*64] | deg[N] | h[N*64]  (~51.6 MB)
  float* agg = (float*)d_ws;
  float* deg = agg + (size_t)N * D_IN;
  float* h   = deg + (size_t)N;

  const int TB = 256;
  const int fill_blocks = 2048;
  const int scat_blocks = (int)(((long)E * 16 + TB - 1) / TB);
  const int deg_blocks  = (E + TB - 1) / TB;
  const int gemm_blocks = ((N / 16) + 7) / 8;   // 8 waves/block, 16 nodes/wave

  // ---- Layer 1 ----
  fill_zero_kernel<<<fill_blocks, TB, 0, stream>>>(agg, (long)N * (D_IN + 1)); // agg + deg
  degree_kernel<<<deg_blocks, TB, 0, stream>>>(dst, deg, E);
  scatter_kernel<<<scat_blocks, TB, 0, stream>>>(x, src, dst, agg, E);
  sage_wmma_kernel<<<gemm_blocks, TB, 0, stream>>>(x, agg, deg, Wl1, Wr1, b1,
                                                   h, N, 64, /*relu=*/1);
  // ---- Layer 2 ----
  fill_zero_kernel<<<fill_blocks, TB, 0, stream>>>(agg, (long)N * D_IN);
  scatter_kernel<<<scat_blocks, TB, 0, stream>>>(h, src, dst, agg, E);
  sage_wmma_kernel<<<gemm_blocks, TB, 0, stream>>>(h, agg, deg, Wl2, Wr2, b2,
                                                   out, N, 32, /*relu=*/0);
}